// GraphAttentionLayer_35287451304381
// MI455X (gfx1250) — compile-verified
//
#include <hip/hip_runtime.h>
#include <hip/hip_bf16.h>

typedef float v2f __attribute__((ext_vector_type(2)));
typedef float v8f __attribute__((ext_vector_type(8)));

#define LRELU_SLOPE 0.01f

// ---------------------------------------------------------------------------
// K1: per-node projections a_src[n] = h[n]·W[0:64], a_dst[n] = h[n]·W[64:128]
// via V_WMMA_F32_16X16X4_F32 (fp32 matrix core). One wave -> 16 nodes.
// Branch-free inner loop: B loads always in-bounds, masked by lane column.
// ---------------------------------------------------------------------------
__global__ __launch_bounds__(256) void gat_proj_wmma(
    const float* __restrict__ h, const float* __restrict__ W,
    float* __restrict__ a_src, float* __restrict__ a_dst, int n_nodes)
{
  const int lane = threadIdx.x & 31;
  const int wave = blockIdx.x * (blockDim.x >> 5) + (threadIdx.x >> 5);
  const int node0 = wave * 16;
  if (node0 >= n_nodes) return;

  const int m    = lane & 15;   // A: row M / B,D: column N
  const int half = lane >> 4;   // lane-half selects K pair

  int row = node0 + m;
  if (row >= n_nodes) row = n_nodes - 1;          // clamp (N%16==0 normally)
  const float* hrow = h + (size_t)row * 64;
  const float* wrow = W + (m & 1) * 64;           // always-valid B address
  const float  bmsk = (m < 2) ? 1.0f : 0.0f;      // zero unused B columns

  v8f acc = {};
  #pragma unroll
  for (int kk = 0; kk < 16; ++kk) {
    const int kbase = kk * 4 + half * 2;          // K for VGPR0; VGPR1 = +1
    v2f afrag = *(const v2f*)(hrow + kbase);      // A[m][kbase..kbase+1]
    v2f braw  = *(const v2f*)(wrow + kbase);
    v2f bfrag;
    bfrag[0] = braw[0] * bmsk;
    bfrag[1] = braw[1] * bmsk;
    acc = __builtin_amdgcn_wmma_f32_16x16x4_f32(
        /*neg_a=*/false, afrag, /*neg_b=*/false, bfrag,
        /*c_mod=*/(short)0, acc, /*reuse_a=*/false, /*reuse_b=*/false);
  }

  // D: lane holds column N=m; VGPR r holds row M = r + 8*half
  if (m < 2) {
    float* outp = (m == 0) ? a_src : a_dst;
    #pragma unroll
    for (int r = 0; r < 8; ++r) {
      int node = node0 + r + 8 * half;
      if (node < n_nodes) outp[node] = acc[r];
    }
  }
}

// ---------------------------------------------------------------------------
// K2: init per-node state
// ---------------------------------------------------------------------------
__global__ void gat_init(float* __restrict__ node_max, unsigned* __restrict__ counts,
                         unsigned* __restrict__ cursor, int n)
{
  int i = blockIdx.x * blockDim.x + threadIdx.x;
  if (i < n) {
    node_max[i] = __int_as_float(0xFF800000);     // -inf
    counts[i] = 0u;
    cursor[i] = 0u;
  }
}

__device__ __forceinline__ void atomicMaxF(float* addr, float v) {
  if (v >= 0.f) atomicMax((int*)addr,          __float_as_int(v));
  else          atomicMin((unsigned int*)addr, __float_as_uint(v));
}

__device__ __forceinline__ float edge_score(const float* a_src, const float* a_dst,
                                            int s, int d, float bias) {
  float e = a_src[s] + a_dst[d] + bias;
  return (e >= 0.f) ? e : LRELU_SLOPE * e;
}

// ---------------------------------------------------------------------------
// K3: per-edge score -> segment max (bit-trick float atomic max) + degree count
// ---------------------------------------------------------------------------
__global__ void gat_edge_max(const float* __restrict__ a_src, const float* __restrict__ a_dst,
                             const int* __restrict__ src, const int* __restrict__ dst,
                             const float* __restrict__ bptr,
                             float* __restrict__ node_max, unsigned* __restrict__ counts,
                             int n_edges)
{
  int i = blockIdx.x * blockDim.x + threadIdx.x;
  if (i >= n_edges) return;
  int s = src[i], d = dst[i];
  float e = edge_score(a_src, a_dst, s, d, bptr[0]);
  atomicMaxF(&node_max[d], e);
  atomicAdd(&counts[d], 1u);
}

// ---------------------------------------------------------------------------
// K4: single-block chunked exclusive scan of counts -> offsets[0..n]
// ---------------------------------------------------------------------------
__global__ __launch_bounds__(1024) void gat_scan(const unsigned* __restrict__ counts,
                                                 unsigned* __restrict__ offsets, int n)
{
  __shared__ unsigned part[1024];
  const int t = threadIdx.x;
  const int chunk = (n + 1023) / 1024;
  const int lo = t * chunk;
  const int hi = (lo + chunk < n) ? lo + chunk : n;

  unsigned s = 0;
  for (int i = lo; i < hi; ++i) s += counts[i];
  part[t] = s;
  __syncthreads();

  for (int off = 1; off < 1024; off <<= 1) {      // Hillis-Steele inclusive scan
    unsigned v = (t >= off) ? part[t - off] : 0u;
    __syncthreads();
    part[t] += v;
    __syncthreads();
  }

  unsigned run = (t == 0) ? 0u : part[t - 1];     // exclusive base for my chunk
  for (int i = lo; i < hi; ++i) { offsets[i] = run; run += counts[i]; }
  if (t == 0) offsets[n] = part[1023];
}

// ---------------------------------------------------------------------------
// K5: scatter edges into destination-CSR buckets (store src + score)
// ---------------------------------------------------------------------------
__global__ void gat_scatter(const float* __restrict__ a_src, const float* __restrict__ a_dst,
                            const int* __restrict__ src, const int* __restrict__ dst,
                            const float* __restrict__ bptr,
                            const unsigned* __restrict__ offsets, unsigned* __restrict__ cursor,
                            int* __restrict__ perm_src, float* __restrict__ perm_e,
                            int n_edges)
{
  int i = blockIdx.x * blockDim.x + threadIdx.x;
  if (i >= n_edges) return;
  int s = src[i], d = dst[i];
  float e = edge_score(a_src, a_dst, s, d, bptr[0]);
  unsigned pos = offsets[d] + atomicAdd(&cursor[d], 1u);
  perm_src[pos] = s;
  perm_e[pos]   = e;
}

// ---------------------------------------------------------------------------
// K6: one wave per node: softmax-normalize + gather-accumulate h[src] rows.
// Pass 2 processes edges in chunks of 32: coalesced (src,e) loads, shfl
// broadcast, prefetch of upcoming h rows; accumulator in 2 VGPRs/lane.
// ---------------------------------------------------------------------------
__global__ __launch_bounds__(256) void gat_aggregate(
    const float* __restrict__ hmat, const unsigned* __restrict__ offsets,
    const float* __restrict__ node_max,
    const int* __restrict__ perm_src, const float* __restrict__ perm_e,
    float* __restrict__ out, int n_nodes)
{
  const int lane = threadIdx.x & 31;
  const int node = blockIdx.x * (blockDim.x >> 5) + (threadIdx.x >> 5);
  if (node >= n_nodes) return;

  const unsigned beg = offsets[node], end = offsets[node + 1];
  float* outrow = out + (size_t)node * 64 + 2 * lane;
  if (beg == end) {                               // empty segment -> zeros
    v2f z = {0.f, 0.f};
    *(v2f*)outrow = z;
    return;
  }

  const float mx = node_max[node];

  // pass 1: sum of exp(e - max), lanes strided over edges, butterfly reduce
  float part = 0.f;
  for (unsigned j = beg + lane; j < end; j += 32)
    part += __expf(perm_e[j] - mx);
  #pragma unroll
  for (int sh = 16; sh > 0; sh >>= 1)
    part += __shfl_xor(part, sh, 32);
  const float inv = 1.0f / part;

  // pass 2: chunk of 32 edges -> coalesced metadata load + shfl broadcast
  float acc0 = 0.f, acc1 = 0.f;
  for (unsigned base = beg; base < end; base += 32) {
    const unsigned j = base + lane;
    const bool ok = (j < end);
    const int   s_l = ok ? perm_src[j] : 0;
    const float e_l = ok ? perm_e[j]   : 0.f;

    // prefetch this chunk's h rows (32 lanes span each 256B row range)
    if (ok) __builtin_prefetch(hmat + (size_t)s_l * 64 + 2 * lane, 0, 0);

    unsigned cnt = end - base;
    if (cnt > 32u) cnt = 32u;
    for (unsigned k = 0; k < cnt; ++k) {
      const int   s = __shfl(s_l, (int)k, 32);
      const float w = __expf(__shfl(e_l, (int)k, 32) - mx) * inv;
      v2f hv = *(const v2f*)(hmat + (size_t)s * 64 + 2 * lane);
      acc0 += w * hv[0];
      acc1 += w * hv[1];
    }
  }
  v2f res = {acc0, acc1};
  *(v2f*)outrow = res;
}

// ---------------------------------------------------------------------------
extern "C" void kernel_launch(void* const* d_in, const int* in_sizes, int n_in,
                              void* d_out, int out_size, void* d_ws, size_t ws_size,
                              hipStream_t stream)
{
  const float* h    = (const float*)d_in[0];
  const int*   src  = (const int*)  d_in[1];
  const int*   dst  = (const int*)  d_in[2];
  const float* W    = (const float*)d_in[3];
  const float* bptr = (const float*)d_in[4];

  const int n_nodes = in_sizes[0] / 64;
  const int n_edges = in_sizes[1];

  // workspace carve-up (all 4-byte types)
  float*    a_src    = (float*)d_ws;                     // N
  float*    a_dst    = a_src + n_nodes;                  // N
  float*    node_max = a_dst + n_nodes;                  // N
  unsigned* counts   = (unsigned*)(node_max + n_nodes);  // N
  unsigned* offsets  = counts + n_nodes;                 // N+1
  unsigned* cursor   = offsets + n_nodes + 1;            // N
  int*      perm_src = (int*)(cursor + n_nodes);         // E
  float*    perm_e   = (float*)(perm_src + n_edges);     // E

  float* out = (float*)d_out;

  // K1: node projections via fp32 WMMA (16 nodes per wave, 8 waves per block)
  {
    int waves  = (n_nodes + 15) / 16;
    int blocks = (waves + 7) / 8;
    gat_proj_wmma<<<blocks, 256, 0, stream>>>(h, W, a_src, a_dst, n_nodes);
  }
  // K2: init
  gat_init<<<(n_nodes + 255) / 256, 256, 0, stream>>>(node_max, counts, cursor, n_nodes);
  // K3: edge scores -> per-dst max + degree
  gat_edge_max<<<(n_edges + 255) / 256, 256, 0, stream>>>(
      a_src, a_dst, src, dst, bptr, node_max, counts, n_edges);
  // K4: exclusive scan -> CSR offsets
  gat_scan<<<1, 1024, 0, stream>>>(counts, offsets, n_nodes);
  // K5: scatter edges into dst buckets
  gat_scatter<<<(n_edges + 255) / 256, 256, 0, stream>>>(
      a_src, a_dst, src, dst, bptr, offsets, cursor, perm_src, perm_e, n_edges);
  // K6: per-node softmax + feature aggregation
  {
    int blocks = (n_nodes + 7) / 8;
    gat_aggregate<<<blocks, 256, 0, stream>>>(
        h, offsets, node_max, perm_src, perm_e, out, n_nodes);
  }
}